// MultiscaleEncoder_74861279969847
// MI455X (gfx1250) — compile-verified
//
#include <hip/hip_runtime.h>
#include <hip/hip_bf16.h>
#include <stdint.h>

#define D 128
#define AS_STRIDE 129                 // +1 dword pad -> bank = (row + k) % 64, conflict-light
#define SMEM_BYTES ((D * AS_STRIDE + D * D) * 4)

typedef __attribute__((ext_vector_type(2))) float v2f;
typedef __attribute__((ext_vector_type(8))) float v8f;

// ---------------------------------------------------------------- copy H = x
__global__ __launch_bounds__(256) void gin_copy(float* __restrict__ dst,
                                                const float* __restrict__ src,
                                                long long n4) {
    long long i = (long long)blockIdx.x * blockDim.x + threadIdx.x;
    if (i < n4) {
        ((float4*)dst)[i] = ((const float4*)src)[i];
    }
}

// ------------------------------------------------ H[dst] += x[src] over edges
// One wave (32 lanes) per edge; lane handles a float4 feature chunk.
__global__ __launch_bounds__(256) void gin_scatter(float* __restrict__ h,
                                                   const float* __restrict__ x,
                                                   const long long* __restrict__ src,
                                                   const long long* __restrict__ dst,
                                                   long long nedges) {
    long long gid = (long long)blockIdx.x * blockDim.x + threadIdx.x;
    long long e = gid >> 5;
    int lane = (int)(gid & 31);
    if (e < nedges) {
        long long s = src[e];
        long long d = dst[e];
        const float4 v = *(const float4*)(x + s * D + lane * 4);
        float* p = h + d * D + lane * 4;
        atomicAdd(p + 0, v.x);
        atomicAdd(p + 1, v.y);
        atomicAdd(p + 2, v.z);
        atomicAdd(p + 3, v.w);
    }
}

// ------------------------------------- out = relu(A @ W + bias), A:[n,128] f32
// Block: 256 threads (8 waves), owns 128 rows. Wave w -> rows [w*16, w*16+16),
// 8 column tiles of 16, K=128 via 32x V_WMMA_F32_16X16X4_F32.
// Safe to run with out == A (block loads all its rows to LDS before storing).
__global__ __launch_bounds__(256) void gin_gemm_relu(const float* __restrict__ A,
                                                     const float* __restrict__ W,
                                                     const float* __restrict__ bias,
                                                     float* __restrict__ out,
                                                     int nrows) {
    extern __shared__ float smem[];
    float* As = smem;                  // [128][AS_STRIDE]
    float* Ws = smem + D * AS_STRIDE;  // [128][128]

    const int tid = threadIdx.x;
    const int wave = tid >> 5;
    const int lane = tid & 31;
    const int rowBase = blockIdx.x * D;

    // Stage A block (zero-fill out-of-range rows) — coalesced float4 loads.
    for (int i = tid; i < D * (D / 4); i += 256) {
        int r = i >> 5;          // 32 float4 per row
        int c4 = (i & 31) * 4;
        int gr = rowBase + r;
        float4 v = make_float4(0.f, 0.f, 0.f, 0.f);
        if (gr < nrows) v = *(const float4*)(A + (long long)gr * D + c4);
        float* p = As + r * AS_STRIDE + c4;
        p[0] = v.x; p[1] = v.y; p[2] = v.z; p[3] = v.w;
    }
    // Stage W (128x128, row-major [k][n]).
    for (int i = tid; i < D * (D / 4); i += 256) {
        int r = i >> 5;
        int c4 = (i & 31) * 4;
        *(float4*)(Ws + r * D + c4) = *(const float4*)(W + r * D + c4);
    }
    __syncthreads();

    const int m = lane & 15;       // M (A-frag) / N (B-frag, C/D column) index
    const int half = lane >> 4;    // lane half selects K pair {0,1} vs {2,3}
    const float* arow = As + (wave * 16 + m) * AS_STRIDE;

    for (int nt = 0; nt < 8; ++nt) {
        const int n0 = nt * 16;
        v8f acc = {};
        for (int kb = 0; kb < D; kb += 4) {
            const int ka = kb + 2 * half;
            v2f a, b;
            a.x = arow[ka];
            a.y = arow[ka + 1];
            b.x = Ws[ka * D + n0 + m];
            b.y = Ws[(ka + 1) * D + n0 + m];
            acc = __builtin_amdgcn_wmma_f32_16x16x4_f32(
                false, a, false, b, (short)0, acc, false, false);
        }
        // Epilogue: bias + ReLU + store per documented 16x16 f32 C/D layout:
        // VGPR v: lanes 0-15 -> M=v, lanes 16-31 -> M=v+8; N = lane%16.
        const float bv = bias[n0 + m];
#pragma unroll
        for (int v = 0; v < 8; ++v) {
            int gr = rowBase + wave * 16 + v + half * 8;
            if (gr < nrows) {
                float val = acc[v] + bv;
                out[(long long)gr * D + n0 + m] = fmaxf(val, 0.f);
            }
        }
    }
}

// ---------------------------------------------------------------- launcher
extern "C" void kernel_launch(void* const* d_in, const int* in_sizes, int n_in,
                              void* d_out, int out_size, void* d_ws, size_t ws_size,
                              hipStream_t stream) {
    const float* x       = (const float*)d_in[0];
    const long long* ei  = (const long long*)d_in[1];   // int64 [2, E]
    const float* W1      = (const float*)d_in[2];       // [L,128,128]
    const float* b1      = (const float*)d_in[3];       // [L,128]
    const float* W2      = (const float*)d_in[4];
    const float* b2      = (const float*)d_in[5];
    float* out = (float*)d_out;
    float* H   = (float*)d_ws;                          // [N,128] scratch

    const int N = in_sizes[0] / D;
    const long long E = (long long)in_sizes[1] / 2;
    const int L = in_sizes[2] / (D * D);
    const long long ND = (long long)N * D;

    const long long n4 = ND / 4;
    const int copyBlocks = (int)((n4 + 255) / 256);
    const long long scatterThreads = E * 32;
    const int scatterBlocks = (int)((scatterThreads + 255) / 256);
    const int gemmBlocks = (N + D - 1) / D;

    const long long* srcIdx = ei;
    const long long* dstIdx = ei + E;

    for (int l = 0; l < L; ++l) {
        const float* xin = (l == 0) ? x : out + (long long)(l - 1) * ND;

        // H = x
        gin_copy<<<copyBlocks, 256, 0, stream>>>(H, xin, n4);
        // H[dst] += x[src]
        gin_scatter<<<scatterBlocks, 256, 0, stream>>>(H, xin, srcIdx, dstIdx, E);
        // H = relu(H @ W1[l] + b1[l])   (in place)
        gin_gemm_relu<<<gemmBlocks, 256, SMEM_BYTES, stream>>>(
            H, W1 + (long long)l * D * D, b1 + (long long)l * D, H, N);
        // out[l] = relu(H @ W2[l] + b2[l])
        gin_gemm_relu<<<gemmBlocks, 256, SMEM_BYTES, stream>>>(
            H, W2 + (long long)l * D * D, b2 + (long long)l * D,
            out + (long long)l * ND, N);
    }
}